// PolynomialMoE_19112604467579
// MI455X (gfx1250) — compile-verified
//
#include <hip/hip_runtime.h>
#include <hip/hip_bf16.h>

typedef __attribute__((ext_vector_type(2))) float v2f;
typedef __attribute__((ext_vector_type(8))) float v8f;

#define NTOK   1048576
#define DIM    2
#define HID    64
#define NEXP   4

// ---- d_ws layout (bytes) ----
// W1S : float4[4*16*32]            @ 0        (32 KB)  layer1 frag table
// B1S : float2[4*16*32]            @ 32768    (16 KB)
// W2S : float2[4*16*4*32]          @ 49152    (64 KB)  WMMA B-fragment table
// meta: int[32]                    @ 114688
// best: u8[NTOK]                   @ 115712   (1 MB)
// bucket: u32[NTOK]                @ 1164288  (4 MB)
#define WS_W1S    0
#define WS_B1S    32768
#define WS_W2S    49152
#define WS_META   114688
#define WS_BEST   115712
#define WS_BUCKET 1164288

// meta indices
#define M_CNT      0   // [4] per-expert token counts
#define M_SCAT     4   // [4] scatter cursors (consumed by pass 3)
#define M_SEGSTART 8   // [4] bucket segment starts
#define M_TILEOFF  16  // [5] per-expert tile offsets + total

// ---------------- pass -1: zero the atomic counters ----------------
__global__ void moe_zero_meta(int* meta) {
    if (threadIdx.x < 4) meta[M_CNT + threadIdx.x] = 0;
}

// ---------------- pass 0: weight swizzle into fragment tables ----------------
// A-frag layout (V_WMMA_F32_16X16X4_F32): lane L holds M=L%16; hi=L/16 selects
// K pair {4c+2hi, 4c+2hi+1} across the 2 VGPRs. B-frag mirrors with N=L%16.
__global__ __launch_bounds__(256) void moe_swizzle(const float* __restrict__ w1,
                                                   const float* __restrict__ b1,
                                                   const float* __restrict__ w2,
                                                   float* __restrict__ W1S,
                                                   float* __restrict__ B1S,
                                                   float* __restrict__ W2S) {
    int i = blockIdx.x * 256 + threadIdx.x;
    if (i < NEXP * 16 * 4 * 32) {  // 8192 B-fragment float2 entries for w2
        int lane = i & 31, t = (i >> 5) & 3, c = (i >> 7) & 15, e = i >> 11;
        int hi = lane >> 4, n16 = lane & 15;
        int k0 = 4 * c + 2 * hi, nn = 16 * t + n16;
        ((float2*)W2S)[i] = make_float2(w2[(e * HID + k0) * HID + nn],
                                        w2[(e * HID + k0 + 1) * HID + nn]);
    }
    if (i < NEXP * 16 * 32) {      // 2048 layer-1 entries
        int lane = i & 31, c = (i >> 5) & 15, e = i >> 9;
        int hi = lane >> 4;
        int j0 = 4 * c + 2 * hi;
        ((float4*)W1S)[i] = make_float4(w1[(e * DIM + 0) * HID + j0],
                                        w1[(e * DIM + 1) * HID + j0],
                                        w1[(e * DIM + 0) * HID + j0 + 1],
                                        w1[(e * DIM + 1) * HID + j0 + 1]);
        ((float2*)B1S)[i] = make_float2(b1[e * HID + j0], b1[e * HID + j0 + 1]);
    }
}

// ---------------- pass 1: router logits + argmax + histogram ----------------
__global__ __launch_bounds__(256) void moe_router(const float* __restrict__ x,
                                                  const float* __restrict__ rW,
                                                  const float* __restrict__ rb,
                                                  float* __restrict__ logitsOut,
                                                  unsigned char* __restrict__ best,
                                                  int* __restrict__ meta) {
    __shared__ int cnt[4];
    __shared__ float srw[8], srb[4];
    int tid = threadIdx.x;
    if (tid < 4) cnt[tid] = 0;
    if (tid < 8) srw[tid] = rW[tid];
    if (tid < 4) srb[tid] = rb[tid];
    __syncthreads();
    int n = blockIdx.x * 256 + tid;
    float2 xv = ((const float2*)x)[n];
    float l[4];
#pragma unroll
    for (int e = 0; e < 4; ++e)
        l[e] = fmaf(xv.x, srw[e], fmaf(xv.y, srw[4 + e], srb[e]));
    ((float4*)logitsOut)[n] = make_float4(l[0], l[1], l[2], l[3]);
    int b = 0; float bv = l[0];
#pragma unroll
    for (int e = 1; e < 4; ++e) if (l[e] > bv) { bv = l[e]; b = e; }  // first-max tie rule
    best[n] = (unsigned char)b;
    atomicAdd(&cnt[b], 1);
    __syncthreads();
    if (tid < 4) atomicAdd(&meta[M_CNT + tid], cnt[tid]);
}

// ---------------- pass 2: prefix sums (tiny) ----------------
__global__ void moe_offsets(int* meta) {
    if (threadIdx.x == 0 && blockIdx.x == 0) {
        int base = 0, tbase = 0;
        for (int e = 0; e < 4; ++e) {
            int c = meta[M_CNT + e];
            meta[M_SEGSTART + e] = base;
            meta[M_SCAT + e] = base;
            meta[M_TILEOFF + e] = tbase;
            base += c;
            tbase += (c + 15) >> 4;
        }
        meta[M_TILEOFF + 4] = tbase;
    }
}

// ---------------- pass 3: bucket token ids by expert ----------------
__global__ __launch_bounds__(256) void moe_scatter(const unsigned char* __restrict__ best,
                                                   int* __restrict__ meta,
                                                   unsigned* __restrict__ bucket) {
    __shared__ int cnt[4];
    __shared__ int base[4];
    int tid = threadIdx.x;
    if (tid < 4) cnt[tid] = 0;
    __syncthreads();
    int n = blockIdx.x * 256 + tid;
    int e = best[n];
    int rank = atomicAdd(&cnt[e], 1);
    __syncthreads();
    if (tid < 4) base[tid] = atomicAdd(&meta[M_SCAT + tid], cnt[tid]);
    __syncthreads();
    bucket[base[e] + rank] = (unsigned)n;
}

// ---------------- pass 4: expert MLP, one wave32 per 16-token tile ----------------
__global__ __launch_bounds__(256) void moe_main(const float* __restrict__ x,
                                                const float* __restrict__ b2,
                                                const float* __restrict__ w3,
                                                const float* __restrict__ b3,
                                                const float* __restrict__ W1S,
                                                const float* __restrict__ B1S,
                                                const float* __restrict__ W2S,
                                                const int* __restrict__ meta,
                                                const unsigned* __restrict__ bucket,
                                                float* __restrict__ out) {
    __shared__ float w3s[NEXP * HID * 2];   // 2 KB
    __shared__ float b3s[NEXP * 2];
    __shared__ float b2s[NEXP * HID];       // 1 KB
    __shared__ float h2s[8][16 * 66];       // 33 KB, stride 66 -> conflict-free
    int tid = threadIdx.x;
    for (int i = tid; i < NEXP * HID * 2; i += 256) w3s[i] = w3[i];
    for (int i = tid; i < NEXP * HID; i += 256) b2s[i] = b2[i];
    if (tid < NEXP * 2) b3s[tid] = b3[tid];
    __syncthreads();

    int wave = tid >> 5, lane = tid & 31;
    int T = blockIdx.x * 8 + wave;
    if (T >= meta[M_TILEOFF + 4]) return;
    int e = 0;
#pragma unroll
    for (int k = 1; k < 4; ++k) if (T >= meta[M_TILEOFF + k]) e = k;
    int tLocal   = T - meta[M_TILEOFF + e];
    int segStart = meta[M_SEGSTART + e];
    int segEnd   = segStart + meta[M_CNT + e];
    int hi = lane >> 4, n16 = lane & 15;

    int slot = segStart + tLocal * 16 + n16;
    unsigned tok = bucket[(slot < segEnd) ? slot : segStart];  // padded lanes replay a valid token
    float2 xv = ((const float2*)x)[tok];

    // ---- layer 1, computed directly in WMMA A-fragment layout ----
    v2f a[16];
    const float4* W1p = (const float4*)W1S + (e * 16) * 32 + lane;
    const float2* B1p = (const float2*)B1S + (e * 16) * 32 + lane;
#pragma unroll
    for (int c = 0; c < 16; ++c) {
        float4 w  = W1p[c * 32];
        float2 bb = B1p[c * 32];
        float h0 = fmaf(xv.x, w.x, fmaf(xv.y, w.y, bb.x));
        float h1 = fmaf(xv.x, w.z, fmaf(xv.y, w.w, bb.y));
        a[c].x = h0 > 0.f ? h0 : 0.f;
        a[c].y = h1 > 0.f ? h1 : 0.f;
    }

    // ---- layer 2: 16 K-chunks x 4 N-tiles of V_WMMA_F32_16X16X4_F32 ----
    v8f acc[4];
#pragma unroll
    for (int t = 0; t < 4; ++t) {
        float bv = b2s[e * HID + t * 16 + n16];
#pragma unroll
        for (int r = 0; r < 8; ++r) acc[t][r] = bv;   // same N per C row -> splat
    }
    const v2f* W2p = (const v2f*)W2S + (e * 16 * 4) * 32 + lane;
#pragma unroll
    for (int c = 0; c < 16; ++c) {
#pragma unroll
        for (int t = 0; t < 4; ++t) {
            v2f bf = W2p[(c * 4 + t) * 32];
            acc[t] = __builtin_amdgcn_wmma_f32_16x16x4_f32(
                false, a[c], false, bf, (short)0, acc[t], false, false);
        }
    }

    // ---- ReLU + bounce h2 through LDS (C/D layout -> row-major) ----
    float* hrow = &h2s[wave][0];
#pragma unroll
    for (int t = 0; t < 4; ++t)
#pragma unroll
        for (int r = 0; r < 8; ++r) {
            float v = acc[t][r];
            hrow[(r + 8 * hi) * 66 + t * 16 + n16] = v > 0.f ? v : 0.f;
        }

    // ---- layer 3 (64 -> 2) on 16 lanes, masked scatter to out ----
    if (lane < 16) {
        int m = lane;
        int s = segStart + tLocal * 16 + m;
        if (s < segEnd) {
            unsigned tk = bucket[s];
            float y0 = b3s[e * 2 + 0], y1 = b3s[e * 2 + 1];
#pragma unroll 8
            for (int k = 0; k < HID; ++k) {
                float hv = hrow[m * 66 + k];
                y0 = fmaf(hv, w3s[(e * HID + k) * 2 + 0], y0);
                y1 = fmaf(hv, w3s[(e * HID + k) * 2 + 1], y1);
            }
            ((float2*)out)[tk] = make_float2(y0, y1);
        }
    }
}

extern "C" void kernel_launch(void* const* d_in, const int* in_sizes, int n_in,
                              void* d_out, int out_size, void* d_ws, size_t ws_size,
                              hipStream_t stream) {
    const float* x  = (const float*)d_in[0];
    const float* rW = (const float*)d_in[1];
    const float* rb = (const float*)d_in[2];
    const float* w1 = (const float*)d_in[3];
    const float* b1 = (const float*)d_in[4];
    const float* w2 = (const float*)d_in[5];
    const float* b2 = (const float*)d_in[6];
    const float* w3 = (const float*)d_in[7];
    const float* b3 = (const float*)d_in[8];

    float* out    = (float*)d_out;                 // [N,2]
    float* logits = (float*)d_out + 2 * NTOK;      // [N,4]

    char* ws = (char*)d_ws;
    float*         W1S    = (float*)(ws + WS_W1S);
    float*         B1S    = (float*)(ws + WS_B1S);
    float*         W2S    = (float*)(ws + WS_W2S);
    int*           meta   = (int*)(ws + WS_META);
    unsigned char* best   = (unsigned char*)(ws + WS_BEST);
    unsigned*      bucket = (unsigned*)(ws + WS_BUCKET);

    moe_zero_meta<<<1, 32, 0, stream>>>(meta);
    moe_swizzle<<<32, 256, 0, stream>>>(w1, b1, w2, W1S, B1S, W2S);
    moe_router<<<NTOK / 256, 256, 0, stream>>>(x, rW, rb, logits, best, meta);
    moe_offsets<<<1, 32, 0, stream>>>(meta);
    moe_scatter<<<NTOK / 256, 256, 0, stream>>>(best, meta, bucket);

    int maxTiles  = NTOK / 16 + NEXP;              // worst-case padding
    int mainBlocks = (maxTiles + 7) / 8;           // 8 waves per 256-thread block
    moe_main<<<mainBlocks, 256, 0, stream>>>(x, b2, w3, b3, W1S, B1S, W2S,
                                             meta, bucket, out);
}